// SLDA_70016556859847
// MI455X (gfx1250) — compile-verified
//
#include <hip/hip_runtime.h>
#include <cstdint>

#define N_COMP  512
#define N_CLASS 10
#define N_TRAIN 32768
#define N_TEST  8192
#define SHRINK  1e-4f
#define NS_ITERS 40
#define KC      32            // K-chunk staged in LDS per double-buffer slot

typedef __attribute__((ext_vector_type(2))) float v2f;
typedef __attribute__((ext_vector_type(8))) float v8f;

// Workspace layout (float offsets)
#define OFF_DIFFS 0
#define OFF_A     (N_TRAIN * N_COMP)                    // 512x512 regularized sigma
#define OFF_X0    (OFF_A  + N_COMP * N_COMP)            // Newton-Schulz ping
#define OFF_X1    (OFF_X0 + N_COMP * N_COMP)            // Newton-Schulz pong
#define OFF_T     (OFF_X1 + N_COMP * N_COMP)            // T = A*X scratch
#define OFF_MEANS (OFF_T  + N_COMP * N_COMP)            // 10x512
#define OFF_WP    (OFF_MEANS + N_CLASS * N_COMP)        // 512x16 (N padded)
#define OFF_B     (OFF_WP + N_COMP * 16)                // 16
#define OFF_SCAL  (OFF_B + 16)                          // 1

// ---------------------------------------------------------------------------
// Kernel 1: per-class sequential scan (true recurrence). Block c owns class c;
// thread d owns dim d; running mean in a register. Stores sqrt(w_t)-scaled
// diffs so sigma reduces to a plain Gram matrix.
// ---------------------------------------------------------------------------
__global__ __launch_bounds__(N_COMP)
void slda_stats(const float* __restrict__ X, const int* __restrict__ y,
                float* __restrict__ diffs, float* __restrict__ means) {
  const int c = blockIdx.x;
  const int d = threadIdx.x;
  float m = 0.0f, cnt = 0.0f;
  for (int t = 0; t < N_TRAIN; ++t) {
    const int yt = y[t];                     // uniform address -> scalar load
    if (yt == c) {
      const float x = X[(size_t)t * N_COMP + d];
      const float diff = x - m;
      const float w = sqrtf((float)t / (float)(t + 1));
      diffs[(size_t)t * N_COMP + d] = diff * w;
      m += diff / (cnt + 1.0f);
      cnt += 1.0f;
    }
  }
  means[c * N_COMP + d] = m;
}

// ---------------------------------------------------------------------------
// Kernel 2: sigma = D^T D * (1-SHRINK)/N + SHRINK*I  via fp32 WMMA with
// double-buffered async global->LDS staging (ASYNCcnt pipeline).
// Workgroup = 512 threads (16 waves) computes a 64x64 output block; each wave
// owns one 16x16 tile. Per 32-K chunk every thread issues one 16-byte
// global_load_async_to_lds_b128 per panel; s_wait_asynccnt 0x2 overlaps the
// next chunk's DMA with this chunk's 8 WMMAs.
// ---------------------------------------------------------------------------
__global__ __launch_bounds__(512)
void slda_gram(const float* __restrict__ D, float* __restrict__ A) {
  __shared__ alignas(16) float Da[2][KC][64];
  __shared__ alignas(16) float Db[2][KC][64];
  const int t    = threadIdx.x;
  const int lane = t & 31;
  const int w    = t >> 5;
  const int TM   = (blockIdx.x >> 3) << 6;       // 8x8 grid of 64x64 blocks
  const int TN   = (blockIdx.x & 7) << 6;
  const int tml  = (w >> 2) << 4;                // wave's tile inside block
  const int tnl  = (w & 3) << 4;
  const int lm   = lane & 15;
  const int kh   = (lane >> 4) << 1;             // K sub-offset (0 or 2)
  // loader mapping: thread t copies 4 floats: LDS row lr, col lc
  const int lr = t >> 4;
  const int lc = (t & 15) << 2;

  // prologue: issue chunk 0 into buffer 0
  {
    const float* ga = D + (size_t)lr * N_COMP + TM + lc;
    const float* gb = D + (size_t)lr * N_COMP + TN + lc;
    const uint32_t la = (uint32_t)(uintptr_t)&Da[0][lr][lc];
    const uint32_t lb = (uint32_t)(uintptr_t)&Db[0][lr][lc];
    asm volatile("global_load_async_to_lds_b128 %0, %1, off"
                 :: "v"(la), "v"(ga) : "memory");
    asm volatile("global_load_async_to_lds_b128 %0, %1, off"
                 :: "v"(lb), "v"(gb) : "memory");
  }

  v8f acc = {};
  int buf = 0;
  for (int ch = 0; ch < N_TRAIN / KC; ++ch) {
    if (ch + 1 < N_TRAIN / KC) {
      const int kn = (ch + 1) * KC;
      const float* ga = D + (size_t)(kn + lr) * N_COMP + TM + lc;
      const float* gb = D + (size_t)(kn + lr) * N_COMP + TN + lc;
      const uint32_t la = (uint32_t)(uintptr_t)&Da[buf ^ 1][lr][lc];
      const uint32_t lb = (uint32_t)(uintptr_t)&Db[buf ^ 1][lr][lc];
      asm volatile("global_load_async_to_lds_b128 %0, %1, off"
                   :: "v"(la), "v"(ga) : "memory");
      asm volatile("global_load_async_to_lds_b128 %0, %1, off"
                   :: "v"(lb), "v"(gb) : "memory");
      asm volatile("s_wait_asynccnt 0x2" ::: "memory");  // current chunk done
    } else {
      asm volatile("s_wait_asynccnt 0x0" ::: "memory");
    }
    __syncthreads();
    #pragma unroll
    for (int kk = 0; kk < KC; kk += 4) {
      v2f a, b;
      a.x = Da[buf][kk + kh][tml + lm];
      a.y = Da[buf][kk + kh + 1][tml + lm];
      b.x = Db[buf][kk + kh][tnl + lm];
      b.y = Db[buf][kk + kh + 1][tnl + lm];
      acc = __builtin_amdgcn_wmma_f32_16x16x4_f32(false, a, false, b,
                                                  (short)0, acc, false, false);
    }
    __syncthreads();
    buf ^= 1;
  }

  const float scale = (1.0f - SHRINK) / (float)N_TRAIN;
  const int rh = (lane >> 4) << 3;
  #pragma unroll
  for (int r = 0; r < 8; ++r) {
    const int row = TM + tml + r + rh;
    const int col = TN + tnl + lm;
    A[row * N_COMP + col] = acc[r] * scale + ((row == col) ? SHRINK : 0.0f);
  }
}

// ---------------------------------------------------------------------------
// Newton-Schulz inverse of SPD A: X0 = A/||A||_1^2 (symmetric => A^T=A),
// X <- 2X - X*(A*X). ||I - A X0|| < 1 guaranteed; 40 iterations covers
// condition numbers up to ~1e5 at fp32 accuracy.
// ---------------------------------------------------------------------------
__global__ __launch_bounds__(512)
void ns_norm(const float* __restrict__ A, float* __restrict__ scal) {
  __shared__ float red[N_COMP];
  const int r = threadIdx.x;
  float s = 0.0f;
  for (int j = 0; j < N_COMP; ++j) s += fabsf(A[r * N_COMP + j]);
  red[r] = s;
  __syncthreads();
  for (int o = N_COMP / 2; o > 0; o >>= 1) {
    if (r < o) red[r] = fmaxf(red[r], red[r + o]);
    __syncthreads();
  }
  if (r == 0) scal[0] = 1.0f / (red[0] * red[0]);
}

__global__ __launch_bounds__(256)
void ns_scale(const float* __restrict__ A, const float* __restrict__ scal,
              float* __restrict__ X) {
  const int i = blockIdx.x * 256 + threadIdx.x;
  X[i] = A[i] * scal[0];
}

// T = A * X   (512x512x512 fp32 WMMA, one wave per 16x16 tile)
__global__ __launch_bounds__(256)
void ns_ax(const float* __restrict__ A, const float* __restrict__ X,
           float* __restrict__ T) {
  const int lane = threadIdx.x & 31;
  const int wid  = blockIdx.x * (blockDim.x >> 5) + (threadIdx.x >> 5);
  const int tm = (wid >> 5) << 4;
  const int tn = (wid & 31) << 4;
  const int lm = lane & 15;
  const int kh = (lane >> 4) << 1;
  v8f acc = {};
  #pragma unroll 4
  for (int k = 0; k < N_COMP; k += 4) {
    const int k0 = k + kh;
    v2f a, b;
    a.x = A[(tm + lm) * N_COMP + k0];
    a.y = A[(tm + lm) * N_COMP + k0 + 1];
    b.x = X[k0 * N_COMP + tn + lm];
    b.y = X[(k0 + 1) * N_COMP + tn + lm];
    acc = __builtin_amdgcn_wmma_f32_16x16x4_f32(false, a, false, b,
                                                (short)0, acc, false, false);
  }
  const int rh = (lane >> 4) << 3;
  #pragma unroll
  for (int r = 0; r < 8; ++r)
    T[(tm + r + rh) * N_COMP + tn + lm] = acc[r];
}

// Xn = 2*X - X*T
__global__ __launch_bounds__(256)
void ns_upd(const float* __restrict__ X, const float* __restrict__ T,
            float* __restrict__ Xn) {
  const int lane = threadIdx.x & 31;
  const int wid  = blockIdx.x * (blockDim.x >> 5) + (threadIdx.x >> 5);
  const int tm = (wid >> 5) << 4;
  const int tn = (wid & 31) << 4;
  const int lm = lane & 15;
  const int kh = (lane >> 4) << 1;
  v8f acc = {};
  #pragma unroll 4
  for (int k = 0; k < N_COMP; k += 4) {
    const int k0 = k + kh;
    v2f a, b;
    a.x = X[(tm + lm) * N_COMP + k0];
    a.y = X[(tm + lm) * N_COMP + k0 + 1];
    b.x = T[k0 * N_COMP + tn + lm];
    b.y = T[(k0 + 1) * N_COMP + tn + lm];
    acc = __builtin_amdgcn_wmma_f32_16x16x4_f32(false, a, false, b,
                                                (short)0, acc, false, false);
  }
  const int rh = (lane >> 4) << 3;
  #pragma unroll
  for (int r = 0; r < 8; ++r) {
    const int row = tm + r + rh;
    const int col = tn + lm;
    Xn[row * N_COMP + col] = 2.0f * X[row * N_COMP + col] - acc[r];
  }
}

// ---------------------------------------------------------------------------
// Wp(512x16) = Lambda(512x512) @ means^T(512x10, zero-padded to 16)
// ---------------------------------------------------------------------------
__global__ __launch_bounds__(128)
void slda_w(const float* __restrict__ Lam, const float* __restrict__ means,
            float* __restrict__ Wp) {
  const int lane = threadIdx.x & 31;
  const int wid  = blockIdx.x * (blockDim.x >> 5) + (threadIdx.x >> 5);
  const int tm = wid << 4;
  const int lm = lane & 15;
  const int kh = (lane >> 4) << 1;
  v8f acc = {};
  #pragma unroll 4
  for (int k = 0; k < N_COMP; k += 4) {
    const int k0 = k + kh;
    v2f a, b;
    a.x = Lam[(tm + lm) * N_COMP + k0];
    a.y = Lam[(tm + lm) * N_COMP + k0 + 1];
    b.x = (lm < N_CLASS) ? means[lm * N_COMP + k0]     : 0.0f;
    b.y = (lm < N_CLASS) ? means[lm * N_COMP + k0 + 1] : 0.0f;
    acc = __builtin_amdgcn_wmma_f32_16x16x4_f32(false, a, false, b,
                                                (short)0, acc, false, false);
  }
  const int rh = (lane >> 4) << 3;
  #pragma unroll
  for (int r = 0; r < 8; ++r)
    Wp[(tm + r + rh) * 16 + lm] = acc[r];
}

// ---------------------------------------------------------------------------
// b[c] = -0.5 * sum_d means[c][d] * Wp[d][c]
// ---------------------------------------------------------------------------
__global__ __launch_bounds__(256)
void slda_bias(const float* __restrict__ means, const float* __restrict__ Wp,
               float* __restrict__ bvec) {
  __shared__ float red[256];
  const int c = blockIdx.x;
  float s = 0.0f;
  for (int d = threadIdx.x; d < N_COMP; d += 256)
    s += means[c * N_COMP + d] * Wp[d * 16 + c];
  red[threadIdx.x] = s;
  __syncthreads();
  for (int o = 128; o > 0; o >>= 1) {
    if (threadIdx.x < o) red[threadIdx.x] += red[threadIdx.x + o];
    __syncthreads();
  }
  if (threadIdx.x == 0) bvec[c] = -0.5f * red[0];
}

// ---------------------------------------------------------------------------
// logits(8192x10) = X_test @ Wp + b   (store only cols < 10)
// ---------------------------------------------------------------------------
__global__ __launch_bounds__(256)
void slda_logits(const float* __restrict__ X, const float* __restrict__ Wp,
                 const float* __restrict__ bvec, float* __restrict__ out) {
  const int lane = threadIdx.x & 31;
  const int wid  = blockIdx.x * (blockDim.x >> 5) + (threadIdx.x >> 5);
  const int tm = wid << 4;
  const int lm = lane & 15;
  const int kh = (lane >> 4) << 1;
  v8f acc = {};
  #pragma unroll 4
  for (int k = 0; k < N_COMP; k += 4) {
    const int k0 = k + kh;
    v2f a, b;
    a.x = X[(size_t)(tm + lm) * N_COMP + k0];
    a.y = X[(size_t)(tm + lm) * N_COMP + k0 + 1];
    b.x = Wp[k0 * 16 + lm];
    b.y = Wp[(k0 + 1) * 16 + lm];
    acc = __builtin_amdgcn_wmma_f32_16x16x4_f32(false, a, false, b,
                                                (short)0, acc, false, false);
  }
  if (lm < N_CLASS) {
    const float bb = bvec[lm];
    const int rh = (lane >> 4) << 3;
    #pragma unroll
    for (int r = 0; r < 8; ++r)
      out[(size_t)(tm + r + rh) * N_CLASS + lm] = acc[r] + bb;
  }
}

// ---------------------------------------------------------------------------
extern "C" void kernel_launch(void* const* d_in, const int* in_sizes, int n_in,
                              void* d_out, int out_size, void* d_ws, size_t ws_size,
                              hipStream_t stream) {
  (void)in_sizes; (void)n_in; (void)out_size; (void)ws_size;
  const float* X_train = (const float*)d_in[0];
  const int*   y_train = (const int*)d_in[1];
  const float* X_test  = (const float*)d_in[2];
  float* ws    = (float*)d_ws;
  float* diffs = ws + OFF_DIFFS;
  float* A     = ws + OFF_A;
  float* X0    = ws + OFF_X0;
  float* X1    = ws + OFF_X1;
  float* T     = ws + OFF_T;
  float* means = ws + OFF_MEANS;
  float* Wp    = ws + OFF_WP;
  float* bvec  = ws + OFF_B;
  float* scal  = ws + OFF_SCAL;
  float* out   = (float*)d_out;

  slda_stats<<<N_CLASS, N_COMP, 0, stream>>>(X_train, y_train, diffs, means);
  slda_gram <<<64, 512, 0, stream>>>(diffs, A);

  // Newton-Schulz: Lambda = pinv(A) for SPD A
  ns_norm <<<1, N_COMP, 0, stream>>>(A, scal);
  ns_scale<<<N_COMP * N_COMP / 256, 256, 0, stream>>>(A, scal, X0);
  float* Xc = X0;
  float* Xn = X1;
  for (int it = 0; it < NS_ITERS; ++it) {
    ns_ax <<<128, 256, 0, stream>>>(A, Xc, T);
    ns_upd<<<128, 256, 0, stream>>>(Xc, T, Xn);
    float* tmp = Xc; Xc = Xn; Xn = tmp;
  }

  slda_w     <<<8, 128, 0, stream>>>(Xc, means, Wp);
  slda_bias  <<<N_CLASS, 256, 0, stream>>>(means, Wp, bvec);
  slda_logits<<<64, 256, 0, stream>>>(X_test, Wp, bvec, out);
}